// SmallRNN_1013612282208
// MI455X (gfx1250) — compile-verified
//
#include <hip/hip_runtime.h>
#include <cmath>

// CDNA5 (gfx1250) wave32 WMMA RNN:
//   per wave: 16 batch rows, H=8 hidden, recurrence via V_WMMA_F32_16X16X4_F32 x2
//   D(lane=N,vgpr=M) -> v_tanh_f32 -> LDS 16x16 tile -> reload as A operand
//   (lane=M, K split by lane-half); single ds_load_2addr_b64 covers both K-halves.

typedef float v2f __attribute__((ext_vector_type(2)));
typedef float v4f __attribute__((ext_vector_type(4)));
typedef float v8f __attribute__((ext_vector_type(8)));

__device__ __forceinline__ float fast_tanh(float v) {
#if __has_builtin(__builtin_amdgcn_tanhf)
    return __builtin_amdgcn_tanhf(v);   // v_tanh_f32 (TRANS op on gfx1250)
#else
    return tanhf(v);
#endif
}

__global__ __launch_bounds__(32)
void rnn_wmma_kernel(const float* __restrict__ x,      // [B, T] (I == 1)
                     const float* __restrict__ w_ih,   // [8]    (I == 1)
                     const float* __restrict__ w_hh,   // [8, 8] row-major
                     const float* __restrict__ b_ih,   // [8]
                     const float* __restrict__ b_hh,   // [8]
                     const float* __restrict__ fc_w,   // [8]    (O == 1)
                     const float* __restrict__ fc_b,   // [1]
                     float* __restrict__ out,          // [B]
                     int T) {
    const int lane = threadIdx.x;      // 0..31 (wave32)
    const int half = lane >> 4;        // lane-half selects K/M split
    const int n    = lane & 15;        // N column (hidden unit) / A-row index
    const int base = blockIdx.x * 16;  // first batch row of this wave's tile

    // -------- per-lane constants (hidden unit n; zero-pad n >= 8) --------
    const bool  vn     = (n < 8);
    const float wih_l  = vn ? w_ih[n] : 0.0f;
    const float bias_l = vn ? (b_ih[n] + b_hh[n]) : 0.0f;

    // B operand = W_hh^T (K x N = 8x16 zero-padded), split into two 4x16 K-halves.
    // B 4x16 f32 layout: VGPR0 = row K=2*half, VGPR1 = row K=2*half+1, N = lane&15.
    // B[k][n] = w_hh[n][k]  (since h_new[n] = sum_k h[k] * w_hh[n][k])
    v2f b_lo, b_hi;
    b_lo.x = vn ? w_hh[n * 8 + 2 * half + 0] : 0.0f;
    b_lo.y = vn ? w_hh[n * 8 + 2 * half + 1] : 0.0f;
    b_hi.x = vn ? w_hh[n * 8 + 2 * half + 4] : 0.0f;
    b_hi.y = vn ? w_hh[n * 8 + 2 * half + 5] : 0.0f;

    // A operand = H tile (M x K = 16x8), two K-halves. A 16x4 f32 layout:
    // lane = M (0..15), VGPR0 = K=2*half, VGPR1 = K=2*half+1.  h0 = 0.
    v2f a_lo = {0.0f, 0.0f};
    v2f a_hi = {0.0f, 0.0f};

    __shared__ float tile[16 * 16];    // h_t, row-major [M][N]

    for (int t = 0; t < T; t += 4) {
        // x for this wave's 8 batch rows (lane-half picks rows 0-7 vs 8-15),
        // 4 timesteps per 16B global_load_b128 (x contiguous in t).
        v4f xq[8];
#pragma unroll
        for (int r = 0; r < 8; ++r) {
            const size_t row = (size_t)(base + r + 8 * half);
            xq[r] = *(const v4f*)(x + row * (size_t)T + t);
        }

#pragma unroll
        for (int s = 0; s < 4; ++s) {
            // C operand in D layout: C[vgpr r][lane] = x_t[row]*w_ih[n] + b[n]
            v8f c;
#pragma unroll
            for (int r = 0; r < 8; ++r)
                c[r] = xq[r][s] * wih_l + bias_l;

            // D = H(16x8) @ W_hh^T(8x16) + C   via two K=4 WMMAs
            v8f d = __builtin_amdgcn_wmma_f32_16x16x4_f32(
                false, a_lo, false, b_lo, (short)0, c, false, false);
            d = __builtin_amdgcn_wmma_f32_16x16x4_f32(
                false, a_hi, false, b_hi, (short)0, d, false, false);

            // tanh, write back row-major: D vgpr r, lane -> (M = r + 8*half, N = n).
            // Padded columns n>=8 stay exactly 0 (B cols & C were 0, tanh(0)=0).
#pragma unroll
            for (int r = 0; r < 8; ++r)
                tile[(r + 8 * half) * 16 + n] = fast_tanh(d[r]);

            __syncthreads();   // single-wave WG: s_wait_dscnt only; orders the
                               // cross-lane stores ahead of the A-operand loads

            // Reload as next step's A operand: lane holds row m = lane&15,
            // both K-halves merge into one ds_load_2addr_b64.
            a_lo.x = tile[n * 16 + 2 * half + 0];
            a_lo.y = tile[n * 16 + 2 * half + 1];
            a_hi.x = tile[n * 16 + 2 * half + 4];
            a_hi.y = tile[n * 16 + 2 * half + 5];

            // NOTE: no second sync needed. The next iteration's ds_stores carry a
            // true data dependence on these loads (loads -> wmma -> tanh -> store),
            // so neither the compiler nor the in-order per-wave LDS pipeline can
            // let the stores pass the loads. Saves one s_wait_dscnt per step on
            // the 2048-step serial chain.
        }
    }

    // -------- epilogue: out[b] = h_T[b,:] . fc_w + fc_b  (O == 1) --------
    // tile still holds h_T (last-iteration stores were ordered by the sync).
    if (lane < 16) {
        float acc = fc_b[0];
#pragma unroll
        for (int k = 0; k < 8; ++k)
            acc += tile[lane * 16 + k] * fc_w[k];
        out[base + lane] = acc;
    }
}

extern "C" void kernel_launch(void* const* d_in, const int* in_sizes, int n_in,
                              void* d_out, int out_size, void* d_ws, size_t ws_size,
                              hipStream_t stream) {
    const float* x    = (const float*)d_in[0];
    const float* w_ih = (const float*)d_in[1];
    const float* w_hh = (const float*)d_in[2];
    const float* b_ih = (const float*)d_in[3];
    const float* b_hh = (const float*)d_in[4];
    const float* fc_w = (const float*)d_in[5];
    const float* fc_b = (const float*)d_in[6];
    float* out = (float*)d_out;

    const int B = out_size;           // O == 1 -> out has B elements
    const int T = in_sizes[0] / B;    // I == 1 -> x has B*T elements

    dim3 grid(B / 16);
    dim3 block(32);                   // one wave32 per workgroup
    rnn_wmma_kernel<<<grid, block, 0, stream>>>(x, w_ih, w_hh, b_ih, b_hh,
                                                fc_w, fc_b, out, T);
}